// StixelLoss_50379966382816
// MI455X (gfx1250) — compile-verified
//
#include <hip/hip_runtime.h>

// Problem constants (from the reference)
#define B_DIM 64
#define H_DIM 256
#define W_DIM 512
#define CH_STRIDE (H_DIM * W_DIM)          // 131072 = 2^17
#define NE_TOTAL (B_DIM * 2 * H_DIM * W_DIM) // 16777216
#define ALPHA_C 1.0f
#define BETA_C 0.001f
#define GAMMA_C 1.0f
#define THRESH_C 0.5f

#define BCE_BLOCKS 1024
#define DENSE_BLOCKS 128   // 64*512 columns / 256 threads
#define BLOCK_T 256

typedef __attribute__((ext_vector_type(2))) float v2f;
typedef __attribute__((ext_vector_type(8))) float v8f;

// Wave32 sum-reduction using the matrix unit:
// A(16x4): lane L contributes v at (M=L%16, K = L<16 ? 0 : 2); K=1,3 are zero.
// B(4x16) = ones  ->  D[m][n] = v_m + v_{m+16}.
// D layout: VGPR r holds rowsum(r) in lanes 0-15, rowsum(r+8) in lanes 16-31,
// so sum of the 8 D VGPRs gives half-sums; one xor-16 shuffle completes it.
// Requires EXEC == all ones (true at every call site here).
__device__ __forceinline__ float wave_reduce_wmma(float v) {
    v2f a = {v, 0.0f};
    v2f b = {1.0f, 1.0f};
    v8f c = {};
    c = __builtin_amdgcn_wmma_f32_16x16x4_f32(
        /*neg_a=*/false, a, /*neg_b=*/false, b,
        /*c_mod=*/(short)0, c, /*reuse_a=*/false, /*reuse_b=*/false);
    float s = ((c[0] + c[1]) + (c[2] + c[3])) + ((c[4] + c[5]) + (c[6] + c[7]));
    s += __shfl_xor(s, 16, 32);
    return s;  // total of all 32 lanes, in every lane
}

// Returns block total in thread 0 (0 elsewhere).
__device__ __forceinline__ float block_reduce(float v) {
    __shared__ float lds[BLOCK_T / 32];
    float w = wave_reduce_wmma(v);
    const int lane = threadIdx.x & 31;
    const int wid  = threadIdx.x >> 5;
    if (lane == 0) lds[wid] = w;
    __syncthreads();
    float s = 0.0f;
    if (threadIdx.x == 0) {
#pragma unroll
        for (int i = 0; i < BLOCK_T / 32; ++i) s += lds[i];
    }
    return s;
}

// ---------------------------------------------------------------------------
// Kernel 1: streaming BCE + num_cuts. float4 loads, grid-stride.
// Emits one weighted partial per block: (-ALPHA/NE)*sum_bce + BETA*cuts.
// ---------------------------------------------------------------------------
__global__ __launch_bounds__(BLOCK_T) void stixel_bce_cuts_kernel(
    const float* __restrict__ x, const float* __restrict__ t,
    float* __restrict__ partial) {
    const int nvec   = NE_TOTAL / 4;                 // 4,194,304 float4s
    const int tid    = blockIdx.x * blockDim.x + threadIdx.x;
    const int stride = gridDim.x * blockDim.x;       // 262,144 -> exactly 16 iters/thread
    const float4* __restrict__ xv = (const float4*)x;
    const float4* __restrict__ tv = (const float4*)t;

    float bce  = 0.0f;   // sum of t*log(x) + (1-t)*log(1-x)
    float cuts = 0.0f;
    for (int i = tid; i < nvec; i += stride) {
        __builtin_prefetch(&xv[i + stride], 0, 0);   // speculative; OOB dropped
        __builtin_prefetch(&tv[i + stride], 0, 0);
        const float4 xi = xv[i];
        const float4 ti = tv[i];
        bce += ti.x * __logf(xi.x) + (1.0f - ti.x) * __logf(1.0f - xi.x);
        bce += ti.y * __logf(xi.y) + (1.0f - ti.y) * __logf(1.0f - xi.y);
        bce += ti.z * __logf(xi.z) + (1.0f - ti.z) * __logf(1.0f - xi.z);
        bce += ti.w * __logf(xi.w) + (1.0f - ti.w) * __logf(1.0f - xi.w);
        // channel bit of flat element 4*i: ((4*i) >> 17) & 1 == (i >> 15) & 1
        if (((i >> 15) & 1) == 0) {                  // channel 0 -> num_cuts
            cuts += (xi.x > THRESH_C) ? 1.0f : 0.0f;
            cuts += (xi.y > THRESH_C) ? 1.0f : 0.0f;
            cuts += (xi.z > THRESH_C) ? 1.0f : 0.0f;
            cuts += (xi.w > THRESH_C) ? 1.0f : 0.0f;
        }
    }
    const float v = (-ALPHA_C / (float)NE_TOTAL) * bce + BETA_C * cuts;
    const float s = block_reduce(v);
    if (threadIdx.x == 0) partial[blockIdx.x] = s;
}

// ---------------------------------------------------------------------------
// Kernel 2: dense prediction loss. One thread per (b,w) column of channel 1.
// Lanes cover consecutive w -> each row step is a 128B coalesced wave read;
// data is L2-resident after kernel 1 (128 MB working set < 192 MB L2).
// ---------------------------------------------------------------------------
__global__ __launch_bounds__(BLOCK_T) void stixel_dense_kernel(
    const float* __restrict__ x, float* __restrict__ partial) {
    const int tid = blockIdx.x * blockDim.x + threadIdx.x;  // 0..32767
    const int b   = tid >> 9;       // / W_DIM
    const int w   = tid & (W_DIM - 1);
    const float* __restrict__ col = x + (size_t)(b * 2 + 1) * CH_STRIDE + w;

    int   prev  = -1;
    float dense = 0.0f;
    // Rows [0, 254]: row 255 never contributes (idx < H-1) and its prev is unused.
#pragma unroll 4
    for (int i = 0; i < H_DIM - 1; ++i) {
        const float v = col[(size_t)i * W_DIM];
        const bool  m = v > THRESH_C;
        const float d = (float)(i - prev);           // >= 1 always
        const float contrib = __fdividef(1.0f, d * d * d);
        dense += (m && (prev > 0)) ? contrib : 0.0f; // prev>0: python truthiness
        prev = m ? i : prev;
    }
    const float s = block_reduce(GAMMA_C * dense);
    if (threadIdx.x == 0) partial[blockIdx.x] = s;
}

// ---------------------------------------------------------------------------
// Kernel 3: deterministic final sum of all weighted partials -> d_out[0].
// ---------------------------------------------------------------------------
__global__ __launch_bounds__(BLOCK_T) void stixel_final_kernel(
    const float* __restrict__ partial, int n, float* __restrict__ out) {
    float s = 0.0f;
    for (int i = threadIdx.x; i < n; i += BLOCK_T) s += partial[i];
    const float tot = block_reduce(s);
    if (threadIdx.x == 0) out[0] = tot;
}

extern "C" void kernel_launch(void* const* d_in, const int* in_sizes, int n_in,
                              void* d_out, int out_size, void* d_ws, size_t ws_size,
                              hipStream_t stream) {
    (void)in_sizes; (void)n_in; (void)out_size; (void)ws_size;
    const float* x = (const float*)d_in[0];   // inputs  [64,2,256,512] f32
    const float* t = (const float*)d_in[1];   // targets [1,64,2,256,512] f32
    float* out = (float*)d_out;               // scalar f32
    float* ws  = (float*)d_ws;                // >= (BCE_BLOCKS+DENSE_BLOCKS)*4 bytes

    stixel_bce_cuts_kernel<<<BCE_BLOCKS, BLOCK_T, 0, stream>>>(x, t, ws);
    stixel_dense_kernel<<<DENSE_BLOCKS, BLOCK_T, 0, stream>>>(x, ws + BCE_BLOCKS);
    stixel_final_kernel<<<1, BLOCK_T, 0, stream>>>(ws, BCE_BLOCKS + DENSE_BLOCKS, out);
}